// ContrasiveLoss_34394098106580
// MI455X (gfx1250) — compile-verified
//
#include <hip/hip_runtime.h>

#define NUM_TRAIN 100000
#define N_CLASS   100
#define BIT       48
#define BATCH     256
#define ALPHA     0.1f
#define MARGIN    96.0f          // 2*BIT

#define NTILE   160              // 100000 = 625 * 160, and 160 = 10 * 16
#define NBLK    (NUM_TRAIN / NTILE)
#define LDSTR   52               // padded LDS row stride (floats) -> conflict-free

typedef __attribute__((ext_vector_type(2))) float v2f;
typedef __attribute__((ext_vector_type(4))) float v4f;
typedef __attribute__((ext_vector_type(8))) float v8f;

// ---- workspace layout (bytes) ----
// [0,      400000) : int   override[NUM_TRAIN]   (-1 = bank row, else batch row)
// [400000, 401024) : int   blabel[BATCH]
// [401024, 402048) : float bnorm[BATCH]
// [402048, 402056) : float accums[2] = { qsum, loss_sum }

__global__ __launch_bounds__(256) void k_init(int* __restrict__ ovr,
                                              float* __restrict__ accums) {
    int gid = blockIdx.x * 256 + threadIdx.x;
    if (gid < NUM_TRAIN) ovr[gid] = -1;
    if (gid < 2) accums[gid] = 0.0f;
}

__global__ __launch_bounds__(256) void k_batch(const float* __restrict__ u,
                                               const float* __restrict__ y,
                                               const int*   __restrict__ ind,
                                               int*   __restrict__ ovr,
                                               int*   __restrict__ blabel,
                                               float* __restrict__ bnorm,
                                               float* __restrict__ qsum) {
    const int b = threadIdx.x;                 // one block of 256 threads
    atomicMax(&ovr[ind[b]], b);                // last occurrence wins

    const float* ur = u + b * BIT;
    float nrm = 0.0f, q = 0.0f;
#pragma unroll
    for (int k = 0; k < BIT; k += 4) {
        v4f v = *(const v4f*)(ur + k);
        nrm += v.x*v.x + v.y*v.y + v.z*v.z + v.w*v.w;
        q += fabsf(1.0f - fabsf(v.x)) + fabsf(1.0f - fabsf(v.y))
           + fabsf(1.0f - fabsf(v.z)) + fabsf(1.0f - fabsf(v.w));
    }
    bnorm[b] = nrm;

    int lab = 0;
    const float* yr = y + b * N_CLASS;
#pragma unroll
    for (int c = 0; c < N_CLASS; c += 4) {     // one-hot -> label, B128 chunks
        v4f v = *(const v4f*)(yr + c);
        if (v.x > 0.5f) lab = c + 0;
        if (v.y > 0.5f) lab = c + 1;
        if (v.z > 0.5f) lab = c + 2;
        if (v.w > 0.5f) lab = c + 3;
    }
    blabel[b] = lab;

    for (int off = 16; off > 0; off >>= 1) q += __shfl_xor(q, off, 32);
    if ((threadIdx.x & 31) == 0) atomicAdd(qsum, q);
}

__global__ __launch_bounds__(256) void k_main(const float* __restrict__ u,
                                              const float* __restrict__ Ubank,
                                              const float* __restrict__ Ybank,
                                              const int*   __restrict__ ovr,
                                              const int*   __restrict__ blabel,
                                              const float* __restrict__ bnorm,
                                              float* __restrict__ loss_sum) {
    __shared__ float sA[BATCH * LDSTR];    // all of u, padded rows
    __shared__ float sU[NTILE * LDSTR];    // override-resolved bank tile
    __shared__ float sUn[NTILE];
    __shared__ int   sUl[NTILE];
    __shared__ float sBn[BATCH];
    __shared__ int   sBl[BATCH];
    __shared__ float sBlockSum;

    const int tid = threadIdx.x;
    const int j0  = blockIdx.x * NTILE;

    if (tid == 0) sBlockSum = 0.0f;

    // ---- async-stage raw U tile: thread t < NTILE owns bank row t ----
    // GLOBAL_LOAD_ASYNC_TO_LDS_B128: VDST = LDS byte offset (low 32 bits of
    // the generic shared address), VADDR = 64-bit global address (GV mode).
    if (tid < NTILE) {
        const unsigned long long g0 =
            (unsigned long long)(uintptr_t)(Ubank + (long)(j0 + tid) * BIT);
        const unsigned l0 = (unsigned)(uintptr_t)(sU + tid * LDSTR);
#pragma unroll
        for (int i = 0; i < BIT / 4; ++i) {    // 12 x B128 per row
            unsigned           lds = l0 + (unsigned)(i * 16);
            unsigned long long gad = g0 + (unsigned long long)(i * 16);
            asm volatile("global_load_async_to_lds_b128 %0, %1, off"
                         :: "v"(lds), "v"(gad) : "memory");
        }
    }

    // ---- stage A = u : thread t owns batch row t (overlaps with async) ----
    {
        const float* src = u + tid * BIT;
        float* dst = sA + tid * LDSTR;
#pragma unroll
        for (int k = 0; k < BIT; k += 4) {
            v4f v = *(const v4f*)(src + k);
            dst[k+0] = v.x; dst[k+1] = v.y; dst[k+2] = v.z; dst[k+3] = v.w;
        }
        sBn[tid] = bnorm[tid];
        sBl[tid] = blabel[tid];
    }

    asm volatile("s_wait_asynccnt 0" ::: "memory");

    // ---- patch scatter-overridden rows, norms and labels (same owner) ----
    if (tid < NTILE) {
        const int j  = j0 + tid;
        const int ov = ovr[j];
        float* dst = sU + tid * LDSTR;
        if (ov >= 0) {                          // virtual scatter U[ind]=u
            const float* src = u + ov * BIT;
#pragma unroll
            for (int k = 0; k < BIT; k += 4) {
                v4f v = *(const v4f*)(src + k);
                dst[k+0] = v.x; dst[k+1] = v.y; dst[k+2] = v.z; dst[k+3] = v.w;
            }
        }
        float nrm = 0.0f;
#pragma unroll
        for (int k = 0; k < BIT; ++k) { float v = dst[k]; nrm += v * v; }
        sUn[tid] = nrm;

        int lab;
        if (ov >= 0) {
            lab = blabel[ov];
        } else {
            lab = 0;
            const float* yr = Ybank + (long)j * N_CLASS;
#pragma unroll
            for (int c = 0; c < N_CLASS; c += 4) {
                v4f v = *(const v4f*)(yr + c);
                if (v.x > 0.5f) lab = c + 0;
                if (v.y > 0.5f) lab = c + 1;
                if (v.z > 0.5f) lab = c + 2;
                if (v.w > 0.5f) lab = c + 3;
            }
        }
        sUl[tid] = lab;
    }
    __syncthreads();

    const int wave = tid >> 5;
    const int lane = tid & 31;
    const int hrow = lane & 15;
    const int hi   = lane >> 4;

    float tsum = 0.0f;

    // 16 mtiles x 10 ntiles = 160 WMMA output tiles, 8 waves
    for (int t = wave; t < 16 * (NTILE / 16); t += 8) {
        const int mt = t / (NTILE / 16);
        const int nt = t % (NTILE / 16);
        const int m0 = mt * 16, n0 = nt * 16;

        // ISA 32-bit A layout: VGPR0 = K{0|2}, VGPR1 = K{1|3} (lanes lo|hi)
        const float* arow = sA + (m0 + hrow) * LDSTR + 2 * hi;
        const float* brow = sU + (n0 + hrow) * LDSTR + 2 * hi;

        v8f acc = {0.f, 0.f, 0.f, 0.f, 0.f, 0.f, 0.f, 0.f};
#pragma unroll
        for (int kb = 0; kb < BIT; kb += 4) {
            v2f a = { arow[kb], arow[kb + 1] };
            v2f b = { brow[kb], brow[kb + 1] };
            acc = __builtin_amdgcn_wmma_f32_16x16x4_f32(
                false, a, false, b, (short)0, acc, false, false);
        }

        // C/D layout: vgpr r, lane -> M = m0 + r + 8*hi, N = n0 + hrow
        const int   n  = n0 + hrow;
        const float un = sUn[n];
        const int   ul = sUl[n];
#pragma unroll
        for (int r = 0; r < 8; ++r) {
            const int m = m0 + r + 8 * hi;
            float dist = sBn[m] + un - 2.0f * acc[r];
            float c = (sBl[m] != ul) ? 0.5f * fmaxf(MARGIN - dist, 0.0f)
                                     : 0.5f * dist;
            tsum += c;
        }
    }

    for (int off = 16; off > 0; off >>= 1) tsum += __shfl_xor(tsum, off, 32);
    if (lane == 0) atomicAdd(&sBlockSum, tsum);
    __syncthreads();
    if (tid == 0) atomicAdd(loss_sum, sBlockSum);
}

__global__ void k_final(const float* __restrict__ accums,
                        float* __restrict__ out) {
    const float loss1 = accums[1] / ((float)BATCH * (float)NUM_TRAIN);
    const float loss2 = ALPHA * accums[0] / ((float)BATCH * (float)BIT);
    out[0] = loss1 + loss2;
}

extern "C" void kernel_launch(void* const* d_in, const int* in_sizes, int n_in,
                              void* d_out, int out_size, void* d_ws, size_t ws_size,
                              hipStream_t stream) {
    (void)in_sizes; (void)n_in; (void)out_size; (void)ws_size;
    const float* u   = (const float*)d_in[0];
    const float* y   = (const float*)d_in[1];
    const int*   ind = (const int*)  d_in[2];
    const float* U   = (const float*)d_in[3];
    const float* Y   = (const float*)d_in[4];

    char* ws = (char*)d_ws;
    int*   ovr    = (int*)  (ws + 0);
    int*   blab   = (int*)  (ws + 400000);
    float* bnorm  = (float*)(ws + 401024);
    float* accums = (float*)(ws + 402048);   // {qsum, loss_sum}

    k_init <<<(NUM_TRAIN + 255) / 256, 256, 0, stream>>>(ovr, accums);
    k_batch<<<1, 256, 0, stream>>>(u, y, ind, ovr, blab, bnorm, accums + 0);
    k_main <<<NBLK, 256, 0, stream>>>(u, U, Y, ovr, blab, bnorm, accums + 1);
    k_final<<<1, 1, 0, stream>>>(accums, (float*)d_out);
}